// ContrastiveLoss_all_40372692582580
// MI455X (gfx1250) — compile-verified
//
#include <hip/hip_runtime.h>

#define BDIM 256
#define DDIM 1024

typedef float v2f __attribute__((ext_vector_type(2)));
typedef float v8f __attribute__((ext_vector_type(8)));

// ---------------------------------------------------------------------------
// Phase 1: gather the six diagonals and fold them into g[i,d] (coalesced).
// One block per row i; 256 threads x float4 = 1024 floats = one row.
// ---------------------------------------------------------------------------
__global__ void diag_gather_kernel(const float* __restrict__ a0, const float* __restrict__ a1,
                                   const float* __restrict__ a2, const float* __restrict__ a3,
                                   const float* __restrict__ a4, const float* __restrict__ a5,
                                   float* __restrict__ g) {
    const int i = blockIdx.x;
    const int t = threadIdx.x;
    const size_t base = (size_t)i * (size_t)(BDIM + 1) * (size_t)DDIM;  // flat offset of A[i,i,0]

    const float4* p0 = (const float4*)(a0 + base);
    const float4* p1 = (const float4*)(a1 + base);
    const float4* p2 = (const float4*)(a2 + base);
    const float4* p3 = (const float4*)(a3 + base);
    const float4* p4 = (const float4*)(a4 + base);
    const float4* p5 = (const float4*)(a5 + base);

    float4 r0 = p0[t], r1 = p1[t], r2 = p2[t];
    float4 r3 = p3[t], r4 = p4[t], r5 = p5[t];

    float4 o;
    o.x = (r0.x + r1.x + r2.x) - (r3.x + r4.x + r5.x);
    o.y = (r0.y + r1.y + r2.y) - (r3.y + r4.y + r5.y);
    o.z = (r0.z + r1.z + r2.z) - (r3.z + r4.z + r5.z);
    o.w = (r0.w + r1.w + r2.w) - (r3.w + r4.w + r5.w);

    ((float4*)(g + (size_t)i * DDIM))[t] = o;
}

// ---------------------------------------------------------------------------
// Phase 2: per-16-row tile, accumulate C += W(16x4) * G^T(4x16) over D with
// V_WMMA_F32_16X16X4_F32; diagonal of C gives the per-row dot products.
// One wave (32 threads) per block -> EXEC all ones for WMMA.
// Per ISA layouts, both A(16x4 f32) and B(4x16 f32) place on lane L the pair
// (row = L&15, d = 4*kt + 2*(L>>4) + {0,1}).
// ---------------------------------------------------------------------------
__global__ void wmma_diag_dot_kernel(const float* __restrict__ g, const float* __restrict__ m,
                                     const float* __restrict__ trm, float* __restrict__ partials) {
    const int lane   = threadIdx.x;      // 0..31
    const int rlocal = lane & 15;
    const int half   = lane >> 4;
    const int row    = blockIdx.x * 16 + rlocal;

    const float2* mrow = (const float2*)(m   + (size_t)row * DDIM);
    const float2* trow = (const float2*)(trm + (size_t)row * DDIM);
    const float2* grow = (const float2*)(g   + (size_t)row * DDIM);

    v8f c = {};
#pragma unroll 4
    for (int kt = 0; kt < DDIM / 4; ++kt) {
        const int idx2 = kt * 2 + half;          // float2 index == (4*kt + 2*half)/2
        float2 mv = mrow[idx2];
        float2 tv = trow[idx2];
        float2 gv = grow[idx2];
        v2f a, b;
        a.x = 0.4f * mv.x + 0.6f * tv.x;
        a.y = 0.4f * mv.y + 0.6f * tv.y;
        b.x = gv.x;
        b.y = gv.y;
        // (neg_a, A, neg_b, B, c_mod, C, reuse_a, reuse_b)
        c = __builtin_amdgcn_wmma_f32_16x16x4_f32(false, a, false, b, (short)0, c,
                                                  false, false);
    }

    // Diagonal of the 16x16 f32 C/D layout:
    //   rows 0..7  -> VGPR j on lane j
    //   rows 8..15 -> VGPR j on lane j+24  (M=j+8, N=lane-16=j+8)
    float s = 0.0f;
#pragma unroll
    for (int j = 0; j < 8; ++j) {
        if (lane == j)      s = c[j];
        if (lane == j + 24) s = c[j];
    }
    s = fmaxf(s, 0.0f);   // relu per row (MARGIN == 0)

    __shared__ float red[32];
    red[lane] = s;
    __syncthreads();
    if (lane == 0) {
        float acc = 0.0f;
#pragma unroll
        for (int j = 0; j < 32; ++j) acc += red[j];   // fixed order: deterministic
        partials[blockIdx.x] = acc;
    }
}

// ---------------------------------------------------------------------------
// Phase 3: fixed-order sum of the 16 tile partials.
// ---------------------------------------------------------------------------
__global__ void final_sum_kernel(const float* __restrict__ partials, float* __restrict__ out) {
    if (threadIdx.x == 0 && blockIdx.x == 0) {
        float acc = 0.0f;
#pragma unroll
        for (int j = 0; j < BDIM / 16; ++j) acc += partials[j];
        out[0] = acc;
    }
}

extern "C" void kernel_launch(void* const* d_in, const int* in_sizes, int n_in,
                              void* d_out, int out_size, void* d_ws, size_t ws_size,
                              hipStream_t stream) {
    (void)in_sizes; (void)n_in; (void)out_size; (void)ws_size;

    const float* a0  = (const float*)d_in[0];  // A_is_t
    const float* a1  = (const float*)d_in[1];  // A_is_t_14
    const float* a2  = (const float*)d_in[2];  // A_is_t_28
    const float* a3  = (const float*)d_in[3];  // A_em_t
    const float* a4  = (const float*)d_in[4];  // A_em_t_14
    const float* a5  = (const float*)d_in[5];  // A_em_t_28
    const float* m   = (const float*)d_in[6];
    const float* trm = (const float*)d_in[7];

    float* ws       = (float*)d_ws;
    float* g        = ws;                              // B*D floats (1 MB)
    float* partials = ws + (size_t)BDIM * DDIM;        // 16 floats
    float* out      = (float*)d_out;

    diag_gather_kernel<<<BDIM, 256, 0, stream>>>(a0, a1, a2, a3, a4, a5, g);
    wmma_diag_dot_kernel<<<BDIM / 16, 32, 0, stream>>>(g, m, trm, partials);
    final_sum_kernel<<<1, 32, 0, stream>>>(partials, out);
}